// my_LRUCell_65893388255693
// MI455X (gfx1250) — compile-verified
//
#include <hip/hip_runtime.h>

// LRU cell, fused: out = epilogue(inputs @ B, states)  on gfx1250 (wave32, WMMA)
//
// Sizes: inputs [8192,1024] f32, B [1024,4096] f32, states [8192,4096] f32,
// phases/log_nus/b_h [2048] f32, out [8192,4096] f32.
//
// GEMM via V_WMMA_F32_16X16X4_F32.
//   WG tile 128(M) x 128(N), 8 waves as 2(M) x 4(N); wave tile 64 x 32
//   (4 M-subtiles x 2 N-subtiles = 8 accumulators, A frags shared across N,
//    B frags shared across M). K staged through LDS in chunks of 16 with a
//   register-staged software pipeline (next chunk's global loads issued
//   before the current chunk's WMMA stream).

typedef __attribute__((ext_vector_type(2))) float v2f;
typedef __attribute__((ext_vector_type(8))) float v8f;

#define NUM_IN    1024
#define NUM_UNITS 2048
#define GN        4096          // GEMM N = 2*NUM_UNITS
#define BATCH_M   8192

#define BM 128                  // workgroup tile M
#define BN 128                  // workgroup tile N
#define BK 16                   // K chunk staged in LDS
#define LDA 20                  // A LDS row stride (floats): 16B-aligned stores, conflict-free b64 reads
#define LDB 136                 // B LDS row stride (floats): 16B-aligned, 2*136%64=16 -> disjoint bank groups

__global__ __launch_bounds__(256, 2)
void lru_fused_wmma(const float* __restrict__ X,      // inputs  [BATCH_M, NUM_IN]
                    const float* __restrict__ S,      // states  [BATCH_M, GN]
                    const float* __restrict__ PH,     // phases  [NUM_UNITS]
                    const float* __restrict__ LN,     // log_nus [NUM_UNITS]
                    const float* __restrict__ Bm,     // B       [NUM_IN, GN]
                    const float* __restrict__ BH,     // b_h     [NUM_UNITS]
                    float* __restrict__ out)          // out     [BATCH_M, GN]
{
    __shared__ float sA[BM * LDA];
    __shared__ float sB[BK * LDB];

    const int tid   = threadIdx.x;
    const int lane  = tid & 31;
    const int wave  = tid >> 5;          // 0..7
    const int waveM = wave >> 2;         // 0..1  (64-row band)
    const int waveN = wave & 3;          // 0..3  (32-col band)
    const int lhalf = lane >> 4;         // 0/1: K-pair select per ISA f32 WMMA layout
    const int l16   = lane & 15;

    const int m0 = blockIdx.y * BM;      // batch-row base
    const int n0 = blockIdx.x * BN;      // gemm-col base

    // staging maps: A = 2 float4/thread, B = 2 float4/thread (coalesced)
    const int aRow = tid >> 2;           // 0..63 (+64 for second half)
    const int aC4  = (tid & 3) << 2;     // 0,4,8,12
    const int bRow = tid >> 5;           // 0..7 (+8 for second half)
    const int bC4  = (tid & 31) << 2;    // 0..124

    const float* Ag = X  + (size_t)m0 * NUM_IN;
    const float* Bg = Bm + (size_t)n0;

    const int fragArow  = waveM * 64 + l16;      // + mt*16
    const int fragBcol0 = waveN * 32 + l16;      // N-subtile 0
    const int fragBcol1 = fragBcol0 + 16;        // N-subtile 1

    v8f acc[4][2];
    #pragma unroll
    for (int i = 0; i < 4; ++i) { acc[i][0] = (v8f){}; acc[i][1] = (v8f){}; }

    // ---- register-staged pipeline ----
    float4 ra0, ra1, rb0, rb1;

    #define LOAD_CHUNK(K0)                                                             \
        ra0 = *(const float4*)(Ag + (size_t)(aRow     ) * NUM_IN + (K0) + aC4);        \
        ra1 = *(const float4*)(Ag + (size_t)(aRow + 64) * NUM_IN + (K0) + aC4);        \
        rb0 = *(const float4*)(Bg + (size_t)((K0) + bRow    ) * GN + bC4);             \
        rb1 = *(const float4*)(Bg + (size_t)((K0) + bRow + 8) * GN + bC4);

    #define STORE_CHUNK()                                                              \
        *(float4*)(&sA[(aRow     ) * LDA + aC4]) = ra0;                                \
        *(float4*)(&sA[(aRow + 64) * LDA + aC4]) = ra1;                                \
        *(float4*)(&sB[(bRow    ) * LDB + bC4]) = rb0;                                 \
        *(float4*)(&sB[(bRow + 8) * LDB + bC4]) = rb1;

    #define COMPUTE_CHUNK()                                                            \
        _Pragma("unroll")                                                              \
        for (int ks = 0; ks < BK; ks += 4) {                                           \
            const int kk = ks + lhalf * 2;                                             \
            v2f bf0, bf1;                                                              \
            bf0.x = sB[(kk    ) * LDB + fragBcol0];                                    \
            bf0.y = sB[(kk + 1) * LDB + fragBcol0];                                    \
            bf1.x = sB[(kk    ) * LDB + fragBcol1];                                    \
            bf1.y = sB[(kk + 1) * LDB + fragBcol1];                                    \
            _Pragma("unroll")                                                          \
            for (int mt = 0; mt < 4; ++mt) {                                           \
                const v2f af = *(const v2f*)(&sA[(fragArow + mt * 16) * LDA + kk]);    \
                acc[mt][0] = __builtin_amdgcn_wmma_f32_16x16x4_f32(                    \
                    false, af, false, bf0, (short)0, acc[mt][0], false, false);        \
                acc[mt][1] = __builtin_amdgcn_wmma_f32_16x16x4_f32(                    \
                    false, af, false, bf1, (short)0, acc[mt][1], false, false);        \
            }                                                                          \
        }

    LOAD_CHUNK(0)
    STORE_CHUNK()
    __syncthreads();

    for (int k0 = 0; k0 + BK < NUM_IN; k0 += BK) {
        LOAD_CHUNK(k0 + BK)          // global loads issued early; latency hidden by WMMA stream
        COMPUTE_CHUNK()
        __syncthreads();             // all waves done reading LDS
        STORE_CHUNK()                // s_wait_loadcnt lands here, long after issue
        __syncthreads();
    }
    COMPUTE_CHUNK()                  // last chunk

    // ---- fused LRU epilogue (no cross-lane traffic) ----
    // GEMM column c: even -> real channel of unit c/2, odd -> imaginary channel.
    // Both of this lane's columns (cg0, cg0+16) share parity.
    const int   cg0  = n0 + fragBcol0;
    const int   cg1  = cg0 + 16;
    const int   odd  = cg0 & 1;
    const int   u0   = cg0 >> 1;
    const int   u1   = cg1 >> 1;
    const float r0   = __expf(-__expf(LN[u0]));
    const float r1   = __expf(-__expf(LN[u1]));
    const float are0 = r0 * __cosf(PH[u0]);
    const float are1 = r1 * __cosf(PH[u1]);
    const float aim0 = r0 * __sinf(PH[u0]);
    const float aim1 = r1 * __sinf(PH[u1]);
    const float bias0 = odd ? 0.0f : BH[u0];       // real bias hits real part only
    const float bias1 = odd ? 0.0f : BH[u1];
    const float sgn0  = odd ? aim0 : -aim0;        // even: -a_im*partner ; odd: +a_im*partner
    const float sgn1  = odd ? aim1 : -aim1;
    const size_t oc0 = (size_t)u0 + (odd ? NUM_UNITS : 0);
    const size_t oc1 = (size_t)u1 + (odd ? NUM_UNITS : 0);

    const int rowBase = m0 + waveM * 64 + lhalf * 8;   // C layout: VGPR r -> M = r + 8*lhalf

    #pragma unroll
    for (int mt = 0; mt < 4; ++mt) {
        #pragma unroll
        for (int r = 0; r < 8; ++r) {
            const int row = rowBase + mt * 16 + r;
            const size_t rb = (size_t)row * GN;
            const float2 sp0 = *(const float2*)(S + rb + (size_t)(cg0 & ~1));  // (re, im)
            const float2 sp1 = *(const float2*)(S + rb + (size_t)(cg1 & ~1));
            const float self0 = odd ? sp0.y : sp0.x, part0 = odd ? sp0.x : sp0.y;
            const float self1 = odd ? sp1.y : sp1.x, part1 = odd ? sp1.x : sp1.y;
            out[rb + oc0] = acc[mt][0][r] + are0 * self0 + sgn0 * part0 + bias0;
            out[rb + oc1] = acc[mt][1][r] + are1 * self1 + sgn1 * part1 + bias1;
        }
    }

    #undef LOAD_CHUNK
    #undef STORE_CHUNK
    #undef COMPUTE_CHUNK
}

extern "C" void kernel_launch(void* const* d_in, const int* in_sizes, int n_in,
                              void* d_out, int out_size, void* d_ws, size_t ws_size,
                              hipStream_t stream) {
    const float* X  = (const float*)d_in[0];  // inputs
    const float* S  = (const float*)d_in[1];  // states
    const float* PH = (const float*)d_in[2];  // phases
    const float* LN = (const float*)d_in[3];  // log_nus
    const float* Bm = (const float*)d_in[4];  // B
    const float* BH = (const float*)d_in[5];  // b_h
    float* out = (float*)d_out;

    dim3 grid(GN / BN, BATCH_M / BM);         // 32 x 64 workgroups
    lru_fused_wmma<<<grid, dim3(256), 0, stream>>>(X, S, PH, LN, Bm, BH, out);
}